// HeteroGAT_45552423141801
// MI455X (gfx1250) — compile-verified
//
#include <hip/hip_runtime.h>
#include <cstddef>
#include <cstdint>

#define HID 128
#define NH 4

typedef __attribute__((ext_vector_type(2))) float v2f;
typedef __attribute__((ext_vector_type(8))) float v8f;

// ---------- float atomic max via int ordering trick ----------
__device__ inline void atomicMaxF(float* addr, float val) {
  if (val >= 0.0f) atomicMax((int*)addr, __float_as_int(val));
  else             atomicMin((unsigned int*)addr, __float_as_uint(val));
}

// ---------- GEMM: Y[N,128] = X[N,K] @ W[K,128] (+ bias) using V_WMMA_F32_16X16X4_F32
// block = 256 threads = 8 waves; block tile = 32 rows x 128 cols; wave = 16-col tile,
// two 16x16 accumulators (row halves). X tile staged in LDS via async global->LDS
// (GLOBAL_LOAD_ASYNC_TO_LDS_B128, ASYNCcnt) on full tiles.
__global__ void __launch_bounds__(256) gemm_wmma_f32(
    const float* __restrict__ X, const float* __restrict__ Wm,
    const float* __restrict__ bias, float* __restrict__ Y, int N, int K) {
  __shared__ float sX[32 * 256];
  const int r0  = blockIdx.x * 32;
  const int tid = threadIdx.x;
  const int kq  = K >> 2;                 // float4s per row
  const int total = 32 * kq;
  if (r0 + 32 <= N) {
    // full tile: async DMA each 16B chunk straight into LDS (per-lane, ASYNCcnt)
    for (int i = tid; i < total; i += 256) {
      int row = i / kq;
      int c4  = (i - row * kq) << 2;
      unsigned lds = (unsigned)(size_t)&sX[row * K + c4];
      unsigned long long ga =
          (unsigned long long)(const void*)(X + (size_t)(r0 + row) * K + c4);
      asm volatile("global_load_async_to_lds_b128 %0, %1, off"
                   :: "v"(lds), "v"(ga) : "memory");
    }
    asm volatile("s_wait_asynccnt 0x0" ::: "memory");
  } else {
    // ragged tail block: zero-pad rows >= N through VGPRs
    for (int i = tid; i < total; i += 256) {
      int row = i / kq;
      int c4  = (i - row * kq) << 2;
      float4 v = make_float4(0.f, 0.f, 0.f, 0.f);
      if (r0 + row < N) v = *(const float4*)(X + (size_t)(r0 + row) * K + c4);
      *(float4*)(&sX[row * K + c4]) = v;
    }
  }
  __syncthreads();

  const int lane = tid & 31;
  const int wv   = tid >> 5;              // 0..7 -> column tile
  const int n0   = wv << 4;
  const int m    = lane & 15;             // A row within tile
  const int kh   = lane >> 4;             // K half select (0 -> K0,1 ; 1 -> K2,3)
  const int n    = lane & 15;             // B/C column within tile

  v8f acc0 = {};
  v8f acc1 = {};
  for (int k = 0; k < K; k += 4) {
    const int kk = k + 2 * kh;
    v2f a0, a1, b;
    a0.x = sX[m * K + kk];          a0.y = sX[m * K + kk + 1];
    a1.x = sX[(m + 16) * K + kk];   a1.y = sX[(m + 16) * K + kk + 1];
    b.x  = Wm[(size_t)kk * HID + n0 + n];
    b.y  = Wm[(size_t)(kk + 1) * HID + n0 + n];
    acc0 = __builtin_amdgcn_wmma_f32_16x16x4_f32(false, a0, false, b, (short)0, acc0, false, false);
    acc1 = __builtin_amdgcn_wmma_f32_16x16x4_f32(false, a1, false, b, (short)0, acc1, false, false);
  }
  const float bb = bias ? bias[n0 + n] : 0.0f;
  for (int v = 0; v < 8; ++v) {
    int row = r0 + v + 8 * kh;            // C/D layout: lanes 0-15 -> M=v, lanes 16-31 -> M=v+8
    if (row < N) Y[(size_t)row * HID + n0 + n] = acc0[v] + bb;
    row += 16;
    if (row < N) Y[(size_t)row * HID + n0 + n] = acc1[v] + bb;
  }
}

// ---------- small utility kernels ----------
__global__ void fill_kernel(float* p, float v, size_t n) {
  size_t i = (size_t)blockIdx.x * 256 + threadIdx.x;
  if (i < n) p[i] = v;
}

__global__ void init_bias2_kernel(float* out, const float* b0, const float* b1, size_t n) {
  size_t i = (size_t)blockIdx.x * 256 + threadIdx.x;
  if (i < n) { int j = (int)(i & (HID - 1)); out[i] = b0[j] + b1[j]; }
}

__global__ void init_bias3_kernel(float* out, const float* b0, const float* b1,
                                  const float* b2, size_t n) {
  size_t i = (size_t)blockIdx.x * 256 + threadIdx.x;
  if (i < n) { int j = (int)(i & (HID - 1)); out[i] = b0[j] + b1[j] + b2[j]; }
}

// a[n,h] = sum_c h[n,h*32+c] * att[h*32+c]; one wave per node, 4 heads reduced in parallel
__global__ void att_reduce_kernel(const float* __restrict__ h, const float* __restrict__ att,
                                  float* __restrict__ out, int N) {
  const int lane = threadIdx.x & 31;
  const int wv   = threadIdx.x >> 5;
  const int nidx = blockIdx.x * 8 + wv;
  if (nidx >= N) return;
  float p[NH];
  for (int k = 0; k < NH; ++k)
    p[k] = h[(size_t)nidx * HID + k * 32 + lane] * att[k * 32 + lane];
  for (int off = 16; off > 0; off >>= 1)
    for (int k = 0; k < NH; ++k) p[k] += __shfl_xor(p[k], off, 32);
  if (lane == 0)
    for (int k = 0; k < NH; ++k) out[(size_t)nidx * NH + k] = p[k];
}

// we_dot[h] = sum_c W_edge[h*32+c] * att_edge[h*32+c]
__global__ void wedot_kernel(const float* We, const float* ae, float* out4) {
  int h = threadIdx.x;                    // 4 threads
  float s = 0.f;
  for (int c = 0; c < 32; ++c) s += We[h * 32 + c] * ae[h * 32 + c];
  out4[h] = s;
}

__global__ void edge_alpha_kernel(const int* __restrict__ src, const int* __restrict__ dst,
                                  const float* __restrict__ as_, const float* __restrict__ ad_,
                                  const float* __restrict__ ea, const float* __restrict__ wedot,
                                  float* __restrict__ alpha, float* __restrict__ amax, int E) {
  int idx = blockIdx.x * 256 + threadIdx.x;
  if (idx >= E * NH) return;
  int e = idx >> 2, h = idx & 3;
  float a = as_[(size_t)src[e] * NH + h] + ad_[(size_t)dst[e] * NH + h];
  if (ea) a += ea[e] * wedot[h];
  a = (a > 0.0f) ? a : 0.2f * a;          // leaky_relu 0.2
  alpha[idx] = a;
  atomicMaxF(&amax[(size_t)dst[e] * NH + h], a);
}

__global__ void edge_exp_kernel(const int* __restrict__ dst, float* __restrict__ alpha,
                                const float* __restrict__ amax, float* __restrict__ den, int E) {
  int idx = blockIdx.x * 256 + threadIdx.x;
  if (idx >= E * NH) return;
  int e = idx >> 2, h = idx & 3;
  float ex = expf(alpha[idx] - amax[(size_t)dst[e] * NH + h]);
  alpha[idx] = ex;
  atomicAdd(&den[(size_t)dst[e] * NH + h], ex);
}

__global__ void edge_scatter_kernel(const int* __restrict__ src, const int* __restrict__ dst,
                                    const float* __restrict__ hs, const float* __restrict__ alpha,
                                    const float* __restrict__ den, float* __restrict__ out, int E) {
  size_t idx = (size_t)blockIdx.x * 256 + threadIdx.x;
  if (idx >= (size_t)E * HID) return;
  int e = (int)(idx >> 7), j = (int)(idx & 127), h = j >> 5;
  int s = src[e], d = dst[e];
  float w = alpha[(size_t)e * NH + h] / (den[(size_t)d * NH + h] + 1e-16f);
  atomicAdd(&out[(size_t)d * HID + j], hs[(size_t)s * HID + j] * w);
}

__global__ void gcn_count_kernel(const int* __restrict__ dst, float* __restrict__ deg, int E) {
  int e = blockIdx.x * 256 + threadIdx.x;
  if (e < E) atomicAdd(&deg[dst[e]], 1.0f);
}

__global__ void rsqrt_kernel(const float* __restrict__ deg, float* __restrict__ dis, int N) {
  int i = blockIdx.x * 256 + threadIdx.x;
  if (i < N) dis[i] = rsqrtf(deg[i]);     // deg >= 1 always (self loop)
}

__global__ void gcn_scatter_kernel(const int* __restrict__ src, const int* __restrict__ dst,
                                   const float* __restrict__ h, const float* __restrict__ dis,
                                   float* __restrict__ out, int E) {
  size_t idx = (size_t)blockIdx.x * 256 + threadIdx.x;
  if (idx >= (size_t)E * HID) return;
  int e = (int)(idx >> 7), j = (int)(idx & 127);
  int s = src[e], d = dst[e];
  atomicAdd(&out[(size_t)d * HID + j], h[(size_t)s * HID + j] * dis[s] * dis[d]);
}

__global__ void gcn_self_kernel(const float* __restrict__ h, const float* __restrict__ dis,
                                float* __restrict__ out, int N) {
  size_t idx = (size_t)blockIdx.x * 256 + threadIdx.x;
  if (idx >= (size_t)N * HID) return;
  int nn = (int)(idx >> 7);
  out[idx] += h[idx] * dis[nn] * dis[nn];
}

__global__ void resid_relu_kernel(const float* __restrict__ raw, const float* __restrict__ x,
                                  float* __restrict__ out, size_t n) {
  size_t i = (size_t)blockIdx.x * 256 + threadIdx.x;
  if (i < n) out[i] = fmaxf(raw[i] + x[i], 0.0f);
}

// =====================================================================
extern "C" void kernel_launch(void* const* d_in, const int* in_sizes, int n_in,
                              void* d_out, int out_size, void* d_ws, size_t ws_size,
                              hipStream_t stream) {
  const float* x_patient   = (const float*)d_in[0];
  const float* x_signature = (const float*)d_in[1];
  const int* ps_src  = (const int*)d_in[2];
  const int* ps_dst  = (const int*)d_in[3];
  const int* fol_src = (const int*)d_in[4];
  const int* fol_dst = (const int*)d_in[5];
  const int* folr_src= (const int*)d_in[6];
  const int* folr_dst= (const int*)d_in[7];
  const int* ss_src  = (const int*)d_in[8];
  const int* ss_dst  = (const int*)d_in[9];
  const float* edge_attr = (const float*)d_in[10];

  const int Np  = in_sizes[0] / 128;
  const int Ns  = in_sizes[1] / 256;
  const int Eps = in_sizes[2];
  const int Ef  = in_sizes[4];
  const int Ess = in_sizes[8];

  // ---- params, flattened in dict-insertion order ----
  int pi = 11;
  auto P = [&]() { return (const float*)d_in[pi++]; };
  struct Gat { const float *Wsrc,*Wdst,*att_s,*att_d,*bias,*Wedge,*att_e; };
  struct Layer { Gat ps, sp, ss; const float *folW,*folb,*folrW,*folrb; };
  const float* in_p_W = P(); const float* in_p_b = P();
  const float* in_s_W = P(); const float* in_s_b = P();
  auto readGat = [&](bool edge) {
    Gat g; g.Wsrc=P(); g.Wdst=P(); g.att_s=P(); g.att_d=P(); g.bias=P();
    if (edge) { g.Wedge=P(); g.att_e=P(); } else { g.Wedge=nullptr; g.att_e=nullptr; }
    return g;
  };
  auto readLayer = [&]() {
    Layer L; L.ps=readGat(true); L.sp=readGat(true); L.ss=readGat(false);
    L.folW=P(); L.folb=P(); L.folrW=P(); L.folrb=P(); return L;
  };
  Layer L1 = readLayer();
  Layer L2 = readLayer();
  const float* out_p_W = P(); const float* out_p_b = P();
  const float* out_s_W = P(); const float* out_s_b = P();

  // ---- workspace carve-up ----
  int Emax = Eps; if (Ef > Emax) Emax = Ef; if (Ess > Emax) Emax = Ess;
  float* wsf = (float*)d_ws;
  size_t off = 0;
  auto alloc = [&](size_t nf) { float* p = wsf + off; off += (nf + 3) & ~(size_t)3; return p; };
  float* xpA  = alloc((size_t)Np * HID);
  float* xsA  = alloc((size_t)Ns * HID);
  float* xpB  = alloc((size_t)Np * HID);
  float* xsB  = alloc((size_t)Ns * HID);
  float* praw = alloc((size_t)Np * HID);
  float* sraw = alloc((size_t)Ns * HID);
  float* hsb  = alloc((size_t)Np * HID);
  float* hdb  = alloc((size_t)Np * HID);
  float* asb  = alloc((size_t)Np * NH);
  float* adb  = alloc((size_t)Np * NH);
  float* alpha= alloc((size_t)Emax * NH);
  float* amax = alloc((size_t)Np * NH);
  float* den  = alloc((size_t)Np * NH);
  float* deg  = alloc((size_t)Np);
  float* dis  = alloc((size_t)Np);
  float* wedot= alloc(4);
  (void)ws_size; (void)n_in; (void)out_size;

  auto gemm = [&](const float* X, const float* Wm, const float* b, float* Y, int N, int K) {
    gemm_wmma_f32<<<dim3((N + 31) / 32), 256, 0, stream>>>(X, Wm, b, Y, N, K);
  };
  auto fill = [&](float* p, float v, size_t n) {
    fill_kernel<<<dim3((unsigned)((n + 255) / 256)), 256, 0, stream>>>(p, v, n);
  };

  auto run_gat = [&](const float* xsrc, int Nsrc, const float* xdst, int Ndst,
                     const int* src, const int* dst, int E, const Gat& g,
                     const float* ea, float* out_acc) {
    gemm(xsrc, g.Wsrc, nullptr, hsb, Nsrc, HID);
    gemm(xdst, g.Wdst, nullptr, hdb, Ndst, HID);
    att_reduce_kernel<<<dim3((Nsrc + 7) / 8), 256, 0, stream>>>(hsb, g.att_s, asb, Nsrc);
    att_reduce_kernel<<<dim3((Ndst + 7) / 8), 256, 0, stream>>>(hdb, g.att_d, adb, Ndst);
    if (ea) wedot_kernel<<<dim3(1), 4, 0, stream>>>(g.Wedge, g.att_e, wedot);
    fill(amax, -__builtin_inff(), (size_t)Ndst * NH);
    fill(den, 0.0f, (size_t)Ndst * NH);
    edge_alpha_kernel<<<dim3((E * NH + 255) / 256), 256, 0, stream>>>(
        src, dst, asb, adb, ea, wedot, alpha, amax, E);
    edge_exp_kernel<<<dim3((E * NH + 255) / 256), 256, 0, stream>>>(dst, alpha, amax, den, E);
    edge_scatter_kernel<<<dim3((unsigned)(((size_t)E * HID + 255) / 256)), 256, 0, stream>>>(
        src, dst, hsb, alpha, den, out_acc, E);
  };

  auto run_gcn = [&](const float* x, const int* src, const int* dst, int E, int N,
                     const float* Wm, float* out_acc) {
    gemm(x, Wm, nullptr, hsb, N, HID);
    fill(deg, 1.0f, (size_t)N);
    gcn_count_kernel<<<dim3((E + 255) / 256), 256, 0, stream>>>(dst, deg, E);
    rsqrt_kernel<<<dim3((N + 255) / 256), 256, 0, stream>>>(deg, dis, N);
    gcn_scatter_kernel<<<dim3((unsigned)(((size_t)E * HID + 255) / 256)), 256, 0, stream>>>(
        src, dst, hsb, dis, out_acc, E);
    gcn_self_kernel<<<dim3((unsigned)(((size_t)N * HID + 255) / 256)), 256, 0, stream>>>(
        hsb, dis, out_acc, N);
  };

  auto run_layer = [&](const Layer& L, const float* xp, const float* xs,
                       float* xp_out, float* xs_out) {
    size_t ns_el = (size_t)Ns * HID, np_el = (size_t)Np * HID;
    // s_raw = GAT_ps + GAT_ss (+ both biases)
    init_bias2_kernel<<<dim3((unsigned)((ns_el + 255) / 256)), 256, 0, stream>>>(
        sraw, L.ps.bias, L.ss.bias, ns_el);
    run_gat(xp, Np, xs, Ns, ps_src, ps_dst, Eps, L.ps, edge_attr, sraw);
    run_gat(xs, Ns, xs, Ns, ss_src, ss_dst, Ess, L.ss, nullptr, sraw);
    // p_raw = GAT_sp + GCN_fol + GCN_folr (+ biases); sp = reversed ps edges
    init_bias3_kernel<<<dim3((unsigned)((np_el + 255) / 256)), 256, 0, stream>>>(
        praw, L.sp.bias, L.folb, L.folrb, np_el);
    run_gat(xs, Ns, xp, Np, ps_dst, ps_src, Eps, L.sp, edge_attr, praw);
    run_gcn(xp, fol_src, fol_dst, Ef, Np, L.folW, praw);
    run_gcn(xp, folr_src, folr_dst, Ef, Np, L.folrW, praw);
    resid_relu_kernel<<<dim3((unsigned)((np_el + 255) / 256)), 256, 0, stream>>>(praw, xp, xp_out, np_el);
    resid_relu_kernel<<<dim3((unsigned)((ns_el + 255) / 256)), 256, 0, stream>>>(sraw, xs, xs_out, ns_el);
  };

  // ---- forward pass ----
  gemm(x_patient,   in_p_W, in_p_b, xpA, Np, 128);
  gemm(x_signature, in_s_W, in_s_b, xsA, Ns, 256);
  run_layer(L1, xpA, xsA, xpB, xsB);
  run_layer(L2, xpB, xsB, xpA, xsA);
  float* out_p = (float*)d_out;
  float* out_s = out_p + (size_t)Np * HID;
  gemm(xpA, out_p_W, out_p_b, out_p, Np, 128);
  gemm(xsA, out_s_W, out_s_b, out_s, Ns, 128);
}